// PSROIPoolhandle_85272280694824
// MI455X (gfx1250) — compile-verified
//
#include <hip/hip_runtime.h>
#include <stdint.h>

#define POOLED   7
#define ODIM     5
#define GS       7
#define CH       (ODIM * GS * GS)   // 245
#define FH       64
#define FW       64
#define SCALE    (1.0f / 16.0f)
#define SLOT     128                 // floats per (pw,c) slot; max window 11*11=121

#define AS1 __attribute__((address_space(1)))
#define AS3 __attribute__((address_space(3)))

// CDNA5 async global->LDS copy (ASYNCcnt-tracked). The builtin's params are
// address-space-qualified int pointers: (AS1 int* gsrc, AS3 int* ldst, imm, imm).
__device__ __forceinline__ void async_copy_f32(const float* g, float* l) {
#if __has_builtin(__builtin_amdgcn_global_load_async_to_lds_b32)
    __builtin_amdgcn_global_load_async_to_lds_b32(
        (AS1 int*)(unsigned long long)g,   // global AS1 ptr (same bit pattern as generic)
        (AS3 int*)l,                       // addrspacecast generic->LDS
        0, 0);
#else
    *l = *g;
#endif
}

__device__ __forceinline__ void async_wait_all() {
#if __has_builtin(__builtin_amdgcn_s_wait_asynccnt)
    __builtin_amdgcn_s_wait_asynccnt(0);
#else
    asm volatile("s_wait_asynccnt 0" ::: "memory");
#endif
}

__global__ __launch_bounds__(256) void psroipool_kernel(
    const float* __restrict__ feat,   // (N, 245, 64, 64)
    const float* __restrict__ rois,   // (R, 5)
    float* __restrict__ out)          // (R, 5, 7, 7)
{
    const int r   = blockIdx.x;
    const int tid = threadIdx.x;

    // ---- ROI geometry: uniform across the block -> scalar loads / SALU ----
    const float* roi = rois + (size_t)r * 5;
    const int   b   = (int)roi[0];
    const float xs  = rintf(roi[1]) * SCALE;
    const float ys  = rintf(roi[2]) * SCALE;
    const float xe  = rintf(roi[3] + 1.0f) * SCALE;
    const float ye  = rintf(roi[4] + 1.0f) * SCALE;
    const float roi_w = fmaxf(xe - xs, 0.1f);
    const float roi_h = fmaxf(ye - ys, 0.1f);
    const float bin_w = roi_w * (1.0f / POOLED);
    const float bin_h = roi_h * (1.0f / POOLED);

    // Per-pw column windows (uniform): wstart/wlen, clipped to [0, FW]
    int ws_[POOLED], wl_[POOLED];
#pragma unroll
    for (int pw = 0; pw < POOLED; ++pw) {
        int w0 = (int)fminf(fmaxf(floorf((float)pw * bin_w + xs), 0.0f), (float)FW);
        int w1 = (int)fminf(fmaxf(ceilf(((float)pw + 1.0f) * bin_w + xs), 0.0f), (float)FW);
        ws_[pw] = w0;
        wl_[pw] = w1 - w0;
    }

    const float* fb = feat + (size_t)b * CH * FH * FW;

    // LDS staging buffer: 7 pw-bins x 5 channels x SLOT floats = 17,920 B
    __shared__ float s_buf[POOLED * ODIM * SLOT];

    for (int ph = 0; ph < POOLED; ++ph) {
        const int hs = (int)fminf(fmaxf(floorf((float)ph * bin_h + ys), 0.0f), (float)FH);
        const int he = (int)fminf(fmaxf(ceilf(((float)ph + 1.0f) * bin_h + ys), 0.0f), (float)FH);
        const int hlen = he - hs;

        // ---- Stage: async-copy all (pw, c) windows of this ph-row into LDS.
        // Consecutive idx -> consecutive ww -> coalesced global addresses.
        for (int pw = 0; pw < POOLED; ++pw) {
            const int w0   = ws_[pw];
            const int wlen = wl_[pw];
            const int per  = hlen * wlen;          // elements per channel window
            const int total = ODIM * per;
            const int chan0 = ph * GS + pw;        // channel for c = 0
            for (int idx = tid; idx < total; idx += 256) {
                const int c   = idx / per;
                const int rem = idx - c * per;
                const int hh  = rem / wlen;
                const int ww  = rem - hh * wlen;
                const float* g = fb
                    + ((size_t)(c * (GS * GS) + chan0) * FH + (hs + hh)) * FW
                    + (w0 + ww);
                float* l = &s_buf[(pw * ODIM + c) * SLOT + rem];
                async_copy_f32(g, l);
            }
        }
        async_wait_all();
        __syncthreads();

        // ---- Reduce: one thread per (c, pw) output of this ph-row ----
        if (tid < ODIM * POOLED) {
            const int pw = tid % POOLED;
            const int c  = tid / POOLED;
            const int wlen = wl_[pw];
            const int cnt  = hlen * wlen;
            float v = 0.0f;
            if (cnt > 0) {
                const float* l = &s_buf[(pw * ODIM + c) * SLOT];
                float s = 0.0f;
                for (int i = 0; i < cnt; ++i) s += l[i];
                v = s / (float)cnt;
            }
            out[(((size_t)r * ODIM + c) * POOLED + ph) * POOLED + pw] = v;
        }
        __syncthreads();   // s_buf reused next ph iteration
    }
}

extern "C" void kernel_launch(void* const* d_in, const int* in_sizes, int n_in,
                              void* d_out, int out_size, void* d_ws, size_t ws_size,
                              hipStream_t stream) {
    const float* feat = (const float*)d_in[0];   // (8, 245, 64, 64) fp32
    const float* rois = (const float*)d_in[1];   // (R, 5) fp32
    float* out = (float*)d_out;                  // (R, 5, 7, 7) fp32
    const int R = in_sizes[1] / 5;
    psroipool_kernel<<<R, 256, 0, stream>>>(feat, rois, out);
}